// EOSEncouragementLoss_9483287789851
// MI455X (gfx1250) — compile-verified
//
#include <hip/hip_runtime.h>
#include <hip/hip_bf16.h>
#include <math.h>

// ---- problem constants (from reference) ----
#define EOS_IDX      2
#define BONUS_WEIGHT 0.1f
#define B_SZ  64
#define T_SZ  128
#define V_SZ  32000
#define VEC4  (V_SZ / 4)      // 8000 float4 per row
#define THREADS 1024          // 32 waves (wave32)

typedef __attribute__((ext_vector_type(2))) float v2f;
typedef __attribute__((ext_vector_type(8))) float v8f;

// Kernel 1: one workgroup per batch row.
// Stage the 128KB logits row into LDS via CDNA5 async global->LDS B128 copies,
// then two LDS passes: block max, block sum of exp(x - max).
// Writes masked eos_prob into ws[b]. Fully deterministic (fixed reduction tree).
__global__ __launch_bounds__(THREADS) void eos_row_kernel(
    const float* __restrict__ logits,
    const int*   __restrict__ target_lengths,
    float*       __restrict__ ws)
{
    extern __shared__ float smem[];          // V_SZ floats + 32 reduction slots
    float* red = smem + V_SZ;

    const int b   = blockIdx.x;
    const int tid = threadIdx.x;
    const int len = target_lengths[b];

    int eos_pos = len - 1;
    eos_pos = eos_pos < 0 ? 0 : (eos_pos > (T_SZ - 1) ? (T_SZ - 1) : eos_pos);

    const float* row = logits + ((size_t)b * T_SZ + (size_t)eos_pos) * (size_t)V_SZ;

    // ---- async stage: global -> LDS, 16B per lane-op, ASYNCcnt tracked ----
    {
        unsigned long long base = (unsigned long long)row;   // SGPR pair (GVS saddr)
        for (int i = tid; i < VEC4; i += THREADS) {
            unsigned int goff = (unsigned int)(i * 16);      // 32-bit vaddr offset
            // low 32 bits of a generic shared pointer == LDS byte offset
            unsigned int loff = (unsigned int)(unsigned long long)(smem + i * 4);
            asm volatile("global_load_async_to_lds_b128 %0, %1, %2"
                         :: "v"(loff), "v"(goff), "s"(base)
                         : "memory");
        }
        asm volatile("s_wait_asynccnt 0x0" ::: "memory");
    }
    __syncthreads();

    const int wave = tid >> 5;
    const int lane = tid & 31;
    const float4* sv = (const float4*)smem;

    // ---- pass A: block max ----
    float lmax = -INFINITY;
    for (int i = tid; i < VEC4; i += THREADS) {
        float4 v = sv[i];
        lmax = fmaxf(lmax, fmaxf(fmaxf(v.x, v.y), fmaxf(v.z, v.w)));
    }
    #pragma unroll
    for (int off = 16; off >= 1; off >>= 1)
        lmax = fmaxf(lmax, __shfl_xor(lmax, off, 32));
    if (lane == 0) red[wave] = lmax;
    __syncthreads();
    float bmax = red[0];
    #pragma unroll
    for (int i = 1; i < 32; ++i) bmax = fmaxf(bmax, red[i]);
    __syncthreads();   // everyone done reading red before it is reused

    // ---- pass B: block sum of exp(x - bmax) ----
    float lsum = 0.0f;
    for (int i = tid; i < VEC4; i += THREADS) {
        float4 v = sv[i];
        lsum += expf(v.x - bmax) + expf(v.y - bmax) +
                expf(v.z - bmax) + expf(v.w - bmax);
    }
    #pragma unroll
    for (int off = 16; off >= 1; off >>= 1)
        lsum += __shfl_xor(lsum, off, 32);
    if (lane == 0) red[wave] = lsum;
    __syncthreads();

    if (tid == 0) {
        float bsum = 0.0f;
        #pragma unroll
        for (int i = 0; i < 32; ++i) bsum += red[i];
        float eos_prob = expf(smem[EOS_IDX] - bmax) / bsum;
        bool valid = (len > 1) && ((len - 1) < T_SZ);
        ws[b] = valid ? eos_prob : 0.0f;
    }
}

// Kernel 2: single wave32, EXEC all ones. Sum 64 workspace values with
// v_wmma_f32_16x16x4_f32 (A = ones 16x4 => D[m][n] = column sums of B; the
// total over all D row-0 entries is layout-independent = sum of all 64 values),
// then a 16-lane shuffle tree, scale, write the scalar loss.
__global__ __launch_bounds__(32) void finalize_kernel(
    const float* __restrict__ ws,
    float*       __restrict__ out)
{
    int lane = threadIdx.x;          // 0..31, no divergence before WMMA
    v2f a;  a[0] = 1.0f; a[1] = 1.0f;                  // uniform ones A-matrix
    v2f bm; bm[0] = ws[lane]; bm[1] = ws[lane + 32];   // 64 values across B
    v8f c = {};
    c = __builtin_amdgcn_wmma_f32_16x16x4_f32(
            /*neg_a=*/false, a, /*neg_b=*/false, bm,
            /*c_mod=*/(short)0, c, /*reuse_a=*/false, /*reuse_b=*/false);

    float t = c[0];                  // lane n (n<16): colsum[n]; lanes 16-31 dup
    t += __shfl_xor(t, 8, 32);
    t += __shfl_xor(t, 4, 32);
    t += __shfl_xor(t, 2, 32);
    t += __shfl_xor(t, 1, 32);       // lane 0: sum over lanes 0..15 = total

    if (lane == 0)
        out[0] = -(t * BONUS_WEIGHT) / (float)B_SZ;
}

extern "C" void kernel_launch(void* const* d_in, const int* in_sizes, int n_in,
                              void* d_out, int out_size, void* d_ws, size_t ws_size,
                              hipStream_t stream) {
    const float* logits  = (const float*)d_in[0];
    // d_in[1] = targets: unused by the reference computation
    const int*   lengths = (const int*)d_in[2];
    float* ws  = (float*)d_ws;     // 64 floats, fully rewritten every call
    float* out = (float*)d_out;    // 1 float

    size_t shbytes = (size_t)(V_SZ + 32) * sizeof(float);  // 128128 B < 320 KB/WGP
    eos_row_kernel<<<B_SZ, THREADS, shbytes, stream>>>(logits, lengths, ws);
    finalize_kernel<<<1, 32, 0, stream>>>(ws, out);
}